// EMPSNLayer_36309653520609
// MI455X (gfx1250) — compile-verified
//
#include <hip/hip_runtime.h>

typedef __attribute__((ext_vector_type(2))) float v2f;
typedef __attribute__((ext_vector_type(8))) float v8f;

#define CH 128
#define MTILES_PER_WAVE 8

// ---------------------------------------------------------------------------
// H = X (Mx128) @ W (128x128), fp32, via V_WMMA_F32_16X16X4_F32.
// One wave (32 lanes) per workgroup. Each wave owns one 16-wide column panel
// of W held in registers (64 VGPRs) and sweeps MTILES_PER_WAVE row-tiles.
// A-operand K-split (lanes 0-15: K=0,1 / lanes 16-31: K=2,3) is folded into
// the load address -> one global_load_b64 per step, no post-load selects.
// ---------------------------------------------------------------------------
__global__ __launch_bounds__(32) void gemm_xw_wmma(
    const float* __restrict__ X, const float* __restrict__ W,
    float* __restrict__ H, int M)
{
  const int lane  = threadIdx.x & 31;
  const int nbase = blockIdx.y << 4;            // which 16-col panel of W
  const int ncol  = nbase + (lane & 15);
  const int hi    = lane >> 4;                  // 0 = lanes 0-15, 1 = lanes 16-31
  const int khalf = hi << 1;                    // K offset 0 or 2 within a step

  // Preload B panel: per K-step s, b[s] = { W[4s+khalf][ncol], W[4s+khalf+1][ncol] }
  v2f breg[32];
#pragma unroll
  for (int s = 0; s < 32; ++s) {
    const int k = (s << 2) + khalf;
    breg[s].x = W[(size_t)k       * CH + ncol];
    breg[s].y = W[(size_t)(k + 1) * CH + ncol];
  }

  const int mrow = lane & 15;                   // A-matrix row within tile
#pragma unroll 1
  for (int it = 0; it < MTILES_PER_WAVE; ++it) {
    const int m0 = (((blockIdx.x << 3) + it) << 4);
    if (m0 >= M) return;                        // M is a multiple of 16
    const float* __restrict__ xbase =
        X + (size_t)(m0 + mrow) * CH + khalf;   // 8B-aligned, lane-group offset
    v8f acc = {};
#pragma unroll
    for (int s = 0; s < 32; ++s) {
      const v2f a = *reinterpret_cast<const v2f*>(xbase + (s << 2));
      acc = __builtin_amdgcn_wmma_f32_16x16x4_f32(
          /*neg_a=*/false, a, /*neg_b=*/false, breg[s],
          /*c_mod=*/(short)0, acc, /*reuse_a=*/false, /*reuse_b=*/false);
    }
    // D layout: VGPR v -> row m0 + hi*8 + v, col ncol
    const int rowb = m0 + (hi << 3);
#pragma unroll
    for (int v = 0; v < 8; ++v)
      H[(size_t)(rowb + v) * CH + ncol] = acc[v];
  }
}

// ---------------------------------------------------------------------------
// out[cols[e], :] += vals[e] * H[rows[e], :]  (segment-sum via f32 atomics)
// One wave per edge; each lane moves a float4 (4 channels).
// ---------------------------------------------------------------------------
__global__ __launch_bounds__(256) void scatter_edges(
    const float* __restrict__ H, const int* __restrict__ rows,
    const int* __restrict__ cols, const float* __restrict__ vals,
    float* __restrict__ out, int nnz)
{
  const int e = blockIdx.x * 8 + (threadIdx.x >> 5);
  if (e >= nnz) return;
  const int lane = threadIdx.x & 31;
  const int   r = rows[e];
  const int   c = cols[e];
  const float v = vals[e];
  const float4 m =
      reinterpret_cast<const float4*>(H + (size_t)r * CH)[lane];
  float* dst = out + (size_t)c * CH + lane * 4;
  atomicAdd(dst + 0, m.x * v);
  atomicAdd(dst + 1, m.y * v);
  atomicAdd(dst + 2, m.z * v);
  atomicAdd(dst + 3, m.w * v);
}

// ---------------------------------------------------------------------------
// Zero-fill (float4 granularity) and elementwise sigmoid.
// ---------------------------------------------------------------------------
__global__ __launch_bounds__(256) void zero_f4(float4* __restrict__ p, long n4)
{
  const long stride = (long)gridDim.x * blockDim.x;
  for (long i = (long)blockIdx.x * blockDim.x + threadIdx.x; i < n4; i += stride)
    p[i] = make_float4(0.f, 0.f, 0.f, 0.f);
}

__global__ __launch_bounds__(256) void sigmoid_f4(float4* __restrict__ p, long n4)
{
  const long stride = (long)gridDim.x * blockDim.x;
  for (long i = (long)blockIdx.x * blockDim.x + threadIdx.x; i < n4; i += stride) {
    float4 q = p[i];
    q.x = 1.f / (1.f + __expf(-q.x));
    q.y = 1.f / (1.f + __expf(-q.y));
    q.z = 1.f / (1.f + __expf(-q.z));
    q.w = 1.f / (1.f + __expf(-q.w));
    p[i] = q;
  }
}

// ---------------------------------------------------------------------------
extern "C" void kernel_launch(void* const* d_in, const int* in_sizes, int n_in,
                              void* d_out, int out_size, void* d_ws, size_t ws_size,
                              hipStream_t stream)
{
  const float* x0  = (const float*)d_in[0];
  const float* x1  = (const float*)d_in[1];
  const float* W1  = (const float*)d_in[3];
  const float* W2  = (const float*)d_in[4];
  const int*   b1r = (const int*)  d_in[5];
  const int*   b1c = (const int*)  d_in[6];
  const float* b1v = (const float*)d_in[7];
  const int*   b2r = (const int*)  d_in[8];
  const int*   b2c = (const int*)  d_in[9];
  const float* b2v = (const float*)d_in[10];

  const int N0   = in_sizes[0] / CH;   // 100000
  const int N1   = in_sizes[1] / CH;   // 400000
  const int NNZ1 = in_sizes[5];        // 800000
  const int NNZ2 = in_sizes[8];        // 900000

  float* out1 = (float*)d_out;
  float* out2 = out1 + (size_t)N1 * CH;
  float* H    = (float*)d_ws;          // reused for h0 then h1 (<= N1*128*4 B)

  // 1) zero both output segments (harness poisons d_out)
  const long n4_out = (long)out_size / 4;
  zero_f4<<<4096, 256, 0, stream>>>((float4*)d_out, n4_out);

  // 2) h0 = x0 @ W1
  {
    const int mtiles = N0 / 16;
    dim3 grid((mtiles + MTILES_PER_WAVE - 1) / MTILES_PER_WAVE, CH / 16);
    gemm_xw_wmma<<<grid, 32, 0, stream>>>(x0, W1, H, N0);
  }

  // 3) out1 += segment_sum(h0[b1_rows] * b1_vals, b1_cols)
  scatter_edges<<<(NNZ1 + 7) / 8, 256, 0, stream>>>(H, b1r, b1c, b1v, out1, NNZ1);

  // 4) h1 = x1 @ W2 (reuses H workspace; stream order serializes vs step 3)
  {
    const int mtiles = N1 / 16;
    dim3 grid((mtiles + MTILES_PER_WAVE - 1) / MTILES_PER_WAVE, CH / 16);
    gemm_xw_wmma<<<grid, 32, 0, stream>>>(x1, W2, H, N1);
  }

  // 5) out2 += segment_sum(h1[b2_rows] * b2_vals, b2_cols)
  scatter_edges<<<(NNZ2 + 7) / 8, 256, 0, stream>>>(H, b2r, b2c, b2v, out2, NNZ2);

  // 6) sigmoid over the whole concatenated output
  sigmoid_f4<<<4096, 256, 0, stream>>>((float4*)d_out, n4_out);
}